// GCNLayer_62397284876370
// MI455X (gfx1250) — compile-verified
//
#include <hip/hip_runtime.h>

typedef __attribute__((ext_vector_type(2))) float v2f;
typedef __attribute__((ext_vector_type(8))) float v8f;

#define IN_F  128
#define OUT_F 128

// ---------------------------------------------------------------------------
// deg[i] = 1.0f  (self-loop from the +eye term)
// ---------------------------------------------------------------------------
__global__ __launch_bounds__(256) void k_deg_init(float* __restrict__ deg, int n) {
    int i = blockIdx.x * blockDim.x + threadIdx.x;
    if (i < n) deg[i] = 1.0f;
}

// ---------------------------------------------------------------------------
// deg[dst] += ew   (column sums of the adjacency)
// ---------------------------------------------------------------------------
__global__ __launch_bounds__(256) void k_deg_scatter(const long long* __restrict__ ei,
                                                     const float* __restrict__ ew,
                                                     float* __restrict__ deg, int e_cnt) {
    int e = blockIdx.x * blockDim.x + threadIdx.x;
    if (e < e_cnt) {
        long long d = ei[(long long)e_cnt + e];   // edge_index[1][e]
        atomicAdd(&deg[d], ew[e]);
    }
}

// ---------------------------------------------------------------------------
// dis[i] = deg[i]^-1/2
// ---------------------------------------------------------------------------
__global__ __launch_bounds__(256) void k_dis(const float* __restrict__ deg,
                                             float* __restrict__ dis, int n) {
    int i = blockIdx.x * blockDim.x + threadIdx.x;
    if (i < n) dis[i] = rsqrtf(deg[i]);
}

// ---------------------------------------------------------------------------
// support = x @ weight  via V_WMMA_F32_16X16X4_F32.
// One wave per 16-row strip; 8 column tiles (full 128 cols) accumulated in
// 8 x v8f.  A-layout: lane&15 = M row, lane>>4 selects K pair {0,1}/{2,3}.
// B-layout: VGPR r holds K=r (lanes 0-15) and K=r+2 (lanes 16-31).
// D-layout: VGPR r holds rows r (lanes 0-15) and r+8 (lanes 16-31).
// ---------------------------------------------------------------------------
__global__ __launch_bounds__(256) void k_support_wmma(const float* __restrict__ x,
                                                      const float* __restrict__ w,
                                                      float* __restrict__ sup, int n) {
    const int lane = threadIdx.x & 31;
    const int wid  = (blockIdx.x * blockDim.x + threadIdx.x) >> 5;
    const int m0   = wid * 16;
    if (m0 >= n) return;
    const int half = lane >> 4;     // 0: K pair {0,1}, 1: K pair {2,3}
    const int l15  = lane & 15;
    const int m    = m0 + l15;

    v8f c[8] = {};                  // c[tn] = 16x16 tile for cols tn*16..tn*16+15
    for (int k = 0; k < IN_F; k += 4) {
        const int kk = k + half * 2;
        v2f a;
        a.x = x[(long long)m * IN_F + kk];
        a.y = x[(long long)m * IN_F + kk + 1];
#pragma unroll
        for (int tn = 0; tn < 8; ++tn) {
            v2f b;
            b.x = w[kk * OUT_F + tn * 16 + l15];
            b.y = w[(kk + 1) * OUT_F + tn * 16 + l15];
            c[tn] = __builtin_amdgcn_wmma_f32_16x16x4_f32(
                /*neg_a=*/false, a, /*neg_b=*/false, b,
                /*c_mod=*/(short)0, c[tn], /*reuse_a=*/false, /*reuse_b=*/false);
        }
    }
#pragma unroll
    for (int tn = 0; tn < 8; ++tn) {
#pragma unroll
        for (int r = 0; r < 8; ++r) {
            int row = m0 + r + half * 8;
            sup[(long long)row * OUT_F + tn * 16 + l15] = c[tn][r];
        }
    }
}

// ---------------------------------------------------------------------------
// out[i,:] = dis[i]^2 * support[i,:]   (diagonal/self-loop term; also
// initializes the poisoned output buffer). One thread per 4 features.
// ---------------------------------------------------------------------------
__global__ __launch_bounds__(256) void k_diag(const float* __restrict__ dis,
                                              const float* __restrict__ sup,
                                              float* __restrict__ out, int n) {
    int t = blockIdx.x * blockDim.x + threadIdx.x;   // n*32 threads
    int i = t >> 5;
    int f4 = t & 31;
    if (i >= n) return;
    float s = dis[i];
    s = s * s;
    const float4* sp = (const float4*)(sup + (long long)i * OUT_F);
    float4 v = sp[f4];
    float4 r = make_float4(s * v.x, s * v.y, s * v.z, s * v.w);
    ((float4*)(out + (long long)i * OUT_F))[f4] = r;
}

// ---------------------------------------------------------------------------
// Edge scatter: one wave per edge, lane handles 4 consecutive features.
// out[src,:] += dis[src]*dis[dst]*ew * support[dst,:]
// support row read is one coalesced 512B burst; out fits in L2 so the
// f32 atomics resolve at the L2 atomic units.
// ---------------------------------------------------------------------------
__global__ __launch_bounds__(256) void k_edge_spmm(const long long* __restrict__ ei,
                                                   const float* __restrict__ ew,
                                                   const float* __restrict__ dis,
                                                   const float* __restrict__ sup,
                                                   float* __restrict__ out, int e_cnt) {
    int wave = (blockIdx.x * blockDim.x + threadIdx.x) >> 5;
    int lane = threadIdx.x & 31;
    if (wave >= e_cnt) return;
    long long s = ei[wave];                        // edge_index[0][e] (row)
    long long d = ei[(long long)e_cnt + wave];     // edge_index[1][e] (col)
    float scale = dis[s] * dis[d] * ew[wave];
    const float4* sp = (const float4*)(sup + d * OUT_F);
    float4 v = sp[lane];
    float* o = out + s * OUT_F + lane * 4;
    atomicAdd(o + 0, scale * v.x);
    atomicAdd(o + 1, scale * v.y);
    atomicAdd(o + 2, scale * v.z);
    atomicAdd(o + 3, scale * v.w);
}

// ---------------------------------------------------------------------------
extern "C" void kernel_launch(void* const* d_in, const int* in_sizes, int n_in,
                              void* d_out, int out_size, void* d_ws, size_t ws_size,
                              hipStream_t stream) {
    const float*     x  = (const float*)d_in[0];
    const long long* ei = (const long long*)d_in[1];   // int64 [2,E]
    const float*     ew = (const float*)d_in[2];
    const float*     w  = (const float*)d_in[3];
    float*           out = (float*)d_out;

    const int n     = in_sizes[0] / IN_F;   // 16384
    const int e_cnt = in_sizes[2];          // 524288

    // workspace: support [n*128] | deg [n] | dis [n]
    float* sup = (float*)d_ws;
    float* deg = sup + (size_t)n * OUT_F;
    float* dis = deg + n;

    // 1) deg[i] = 1
    k_deg_init<<<(n + 255) / 256, 256, 0, stream>>>(deg, n);

    // 2) support = x @ weight  (WMMA f32)
    {
        int waves  = n / 16;                 // 1024 row strips
        int blocks = (waves * 32 + 255) / 256;
        k_support_wmma<<<blocks, 256, 0, stream>>>(x, w, sup, n);
    }

    // 3) deg[dst] += ew
    k_deg_scatter<<<(e_cnt + 255) / 256, 256, 0, stream>>>(ei, ew, deg, e_cnt);

    // 4) dis = rsqrt(deg)
    k_dis<<<(n + 255) / 256, 256, 0, stream>>>(deg, dis, n);

    // 5) out = diag term (initializes out)
    k_diag<<<(n * 32 + 255) / 256, 256, 0, stream>>>(dis, sup, out, n);

    // 6) edge scatter accumulate
    {
        long long threads = (long long)e_cnt * 32;
        int blocks = (int)((threads + 255) / 256);
        k_edge_spmm<<<blocks, 256, 0, stream>>>(ei, ew, dis, sup, out, e_cnt);
    }
}